// ChamferLoss_63445256896851
// MI455X (gfx1250) — compile-verified
//
#include <hip/hip_runtime.h>
#include <hip/hip_bf16.h>
#include <math.h>

typedef __attribute__((ext_vector_type(2))) float v2f;
typedef __attribute__((ext_vector_type(8))) float v8f;

#define NB 8
#define NV 4096
#define NTILES (NV / 16)   // 256 tiles of 16 points

// ---------------------------------------------------------------------------
// Init: column-min array to +inf (bit pattern 0x7F800000, valid for uint-min).
// ---------------------------------------------------------------------------
__global__ void chamfer_init(float* __restrict__ ymin) {
    int i = blockIdx.x * blockDim.x + threadIdx.x;
    if (i < NB * NV) ymin[i] = __builtin_inff();
}

// ---------------------------------------------------------------------------
// Main: one wave32 per (batch, 16-row x-tile). Sweeps all 256 y-tiles.
// Gram tile via V_WMMA_F32_16X16X4_F32 (K=3 padded to 4, fp32 throughout).
// Row mins kept in registers (squared domain; sqrt hoisted out — monotone).
// Column mins pushed with order-independent uint atomicMin from all lanes
// (no cross-half shuffle, no EXEC branch in the hot loop).
// Y-tile loads are software-pipelined so the WMMA epilogue hides latency.
// ---------------------------------------------------------------------------
__global__ __launch_bounds__(256) void chamfer_main(const float* __restrict__ x,
                                                    const float* __restrict__ y,
                                                    float* __restrict__ ymin,
                                                    float* __restrict__ xpart) {
    const int lane = threadIdx.x & 31;
    const int wv   = threadIdx.x >> 5;            // wave in block: 0..7
    const int xt   = blockIdx.x * 8 + wv;         // x tile: 0..255
    const int n    = blockIdx.y;                  // batch
    const int hi   = lane >> 4;                   // half-wave select (K-split)
    const int p    = lane & 15;                   // point-in-tile (M or N index)

    // --- A fragment: x tile (16 points x 3 comps, K padded to 4) ---
    const float* xb = x + ((size_t)n * NV + (size_t)xt * 16) * 3;
    float xc0 = xb[p * 3 + 0];
    float xc1 = xb[p * 3 + 1];
    float xc2 = xb[p * 3 + 2];
    float x2v = xc0 * xc0 + xc1 * xc1 + xc2 * xc2;   // ||x_p||^2
    v2f a;
    a.x = hi ? xc2 : xc0;      // VGPR0: K=0 (lanes 0-15) / K=2 (lanes 16-31)
    a.y = hi ? 0.0f : xc1;     // VGPR1: K=1 / K=3 (zero pad)

    // x2 for the 8 output rows this lane touches (row = r + 8*hi)
    float x2row[8];
#pragma unroll
    for (int r = 0; r < 8; ++r) x2row[r] = __shfl(x2v, r + 8 * hi, 32);

    float rm[8];               // row mins of clamped squared distance
#pragma unroll
    for (int r = 0; r < 8; ++r) rm[r] = __builtin_inff();

    const float* ybase = y + (size_t)n * NV * 3;
    unsigned* yminu = (unsigned*)(ymin + (size_t)n * NV);

    // Software-pipelined y-tile stream: preload tile 0.
    float yc0 = ybase[p * 3 + 0];
    float yc1 = ybase[p * 3 + 1];
    float yc2 = ybase[p * 3 + 2];

    for (int jt = 0; jt < NTILES; ++jt) {
        const float c0 = yc0, c1 = yc1, c2 = yc2;

        // Issue next tile's load early; epilogue below hides its latency.
        const int jn = (jt + 1) & (NTILES - 1);
        const float* yn = ybase + jn * 48;
        yc0 = yn[p * 3 + 0];
        yc1 = yn[p * 3 + 1];
        yc2 = yn[p * 3 + 2];
        // Warm L0/L2 a few tiles ahead.
        __builtin_prefetch(ybase + ((jt + 4) & (NTILES - 1)) * 48 + p * 3, 0, 0);

        float y2v = c0 * c0 + c1 * c1 + c2 * c2;  // ||y_p||^2
        v2f b;
        b.x = hi ? c2 : c0;
        b.y = hi ? 0.0f : c1;

        // 16x16 gram tile: D = A(16x4) * B(4x16)
        v8f acc = {};
        acc = __builtin_amdgcn_wmma_f32_16x16x4_f32(
            /*neg_a=*/false, a, /*neg_b=*/false, b,
            /*c_mod=*/(short)0, acc, /*reuse_a=*/false, /*reuse_b=*/false);

        float cmin = __builtin_inff();
#pragma unroll
        for (int r = 0; r < 8; ++r) {
            // squared distance, clamped (sqrt deferred: monotone)
            float sq = fmaxf(x2row[r] + y2v - 2.0f * acc[r], 0.0f);
            rm[r] = fminf(rm[r], sq);
            cmin  = fminf(cmin, sq);
        }
        // Both half-waves min into the same column slot; uint-min is exact,
        // commutative and idempotent -> deterministic, no shuffle needed.
        atomicMin(&yminu[jt * 16 + p], __float_as_uint(cmin));
    }

    // Row-min reduction across the 16 lanes of each half-wave.
#pragma unroll
    for (int r = 0; r < 8; ++r) {
#pragma unroll
        for (int off = 1; off < 16; off <<= 1)
            rm[r] = fminf(rm[r], __shfl_xor(rm[r], off, 32));
    }
    float s = 0.0f;
#pragma unroll
    for (int r = 0; r < 8; ++r) s += sqrtf(rm[r]);   // rows r (lo) / r+8 (hi)
    s += __shfl_xor(s, 16, 32);                      // all 16 rows of tile
    if (lane == 0) xpart[n * NTILES + xt] = s;       // fixed-order partials
}

// ---------------------------------------------------------------------------
// Finalize: sum sqrt(col-mins) + row-min partials in a deterministic order.
// out = (sum_row_mins + sum_col_mins) / (NB * NV)
// ---------------------------------------------------------------------------
__global__ __launch_bounds__(256) void chamfer_finalize(const float* __restrict__ ymin,
                                                        const float* __restrict__ xpart,
                                                        float* __restrict__ out) {
    __shared__ float red[256];
    float acc = 0.0f;
    for (int i = threadIdx.x; i < NB * NV; i += 256) acc += sqrtf(ymin[i]);
    for (int i = threadIdx.x; i < NB * NTILES; i += 256) acc += xpart[i];
    red[threadIdx.x] = acc;
    __syncthreads();
    for (int s = 128; s > 0; s >>= 1) {
        if ((int)threadIdx.x < s) red[threadIdx.x] += red[threadIdx.x + s];
        __syncthreads();
    }
    if (threadIdx.x == 0) out[0] = red[0] / (float)(NB * NV);
}

extern "C" void kernel_launch(void* const* d_in, const int* in_sizes, int n_in,
                              void* d_out, int out_size, void* d_ws, size_t ws_size,
                              hipStream_t stream) {
    const float* x = (const float*)d_in[0];
    const float* y = (const float*)d_in[1];
    float* out   = (float*)d_out;
    float* ymin  = (float*)d_ws;          // NB*NV floats (col mins, squared)
    float* xpart = ymin + NB * NV;        // NB*NTILES floats (row-min sums)

    chamfer_init<<<(NB * NV + 255) / 256, 256, 0, stream>>>(ymin);

    dim3 grid(NTILES / 8, NB);            // 8 waves per block, 1 x-tile per wave
    chamfer_main<<<grid, 256, 0, stream>>>(x, y, ymin, xpart);

    chamfer_finalize<<<1, 256, 0, stream>>>(ymin, xpart, out);
}